// ContextGNN_90305982365988
// MI455X (gfx1250) — compile-verified
//
#include <hip/hip_runtime.h>
#include <hip/hip_bf16.h>
#include <math.h>

typedef __attribute__((ext_vector_type(2))) float v2f;
typedef __attribute__((ext_vector_type(8))) float v8f;

#define BATCH 256
#define SEQ   128
#define DIM   128
#define HEADS 8
#define KTOP  64
#define EDGES 192   // KTOP + SEQ self loops

#define WMMA_F32(a, b, c) \
    __builtin_amdgcn_wmma_f32_16x16x4_f32(false, (a), false, (b), (short)0, (c), false, false)

// ---------------------------------------------------------------------------
// WMMA f32 GEMM: C[M,N] (+)= A[M,K] @ B[N,K]^T, one wave per 16x64 C tile
// (4 accumulators), K-step 4, software-pipelined fragment loads (all b64).
// Layouts per CDNA5 ISA 7.12.2:
//   A 16x4 : lane m = l&15 (both halves), VGPR0 holds K=2*hi, VGPR1 K=2*hi+1
//   B 4x16 : lane n = l&15, VGPR0 holds K=2*hi row, VGPR1 K=2*hi+1
//   C 16x16: VGPR j -> row j + 8*hi, col = l&15
// ---------------------------------------------------------------------------
__global__ __launch_bounds__(32)
void gemm_nt_f32(const float* __restrict__ A, const float* __restrict__ Bm,
                 float* __restrict__ C, int M, int N, int K,
                 long sA, long sB, long sC, int beta) {
    int lane = threadIdx.x & 31;
    int nt = blockIdx.x, mt = blockIdx.y, b = blockIdx.z;
    const float* Ab = A + (long)b * sA;
    const float* Bb = Bm + (long)b * sB;
    float* Cb = C + (long)b * sC;
    int hi = lane >> 4, lo = lane & 15;
    int m0 = mt * 16, n0 = nt * 64;
    v8f acc0 = {}, acc1 = {}, acc2 = {}, acc3 = {};
    if (beta) {
        for (int j = 0; j < 8; ++j) {
            long r = (long)(m0 + j + 8 * hi) * N + n0 + lo;
            acc0[j] = Cb[r];
            acc1[j] = Cb[r + 16];
            acc2[j] = Cb[r + 32];
            acc3[j] = Cb[r + 48];
        }
    }
    const float* arow = Ab + (long)(m0 + lo) * K + 2 * hi;
    const float* br0  = Bb + (long)(n0 + 0  + lo) * K + 2 * hi;
    const float* br1  = Bb + (long)(n0 + 16 + lo) * K + 2 * hi;
    const float* br2  = Bb + (long)(n0 + 32 + lo) * K + 2 * hi;
    const float* br3  = Bb + (long)(n0 + 48 + lo) * K + 2 * hi;

    // prologue: fragments for K-step 0
    v2f a  = *(const v2f*)arow;
    v2f b0 = *(const v2f*)br0;
    v2f b1 = *(const v2f*)br1;
    v2f b2 = *(const v2f*)br2;
    v2f b3 = *(const v2f*)br3;
    // steady state: issue next K-step's loads before this step's WMMAs
    for (int k0 = 4; k0 < K; k0 += 4) {
        v2f an  = *(const v2f*)(arow + k0);
        v2f b0n = *(const v2f*)(br0 + k0);
        v2f b1n = *(const v2f*)(br1 + k0);
        v2f b2n = *(const v2f*)(br2 + k0);
        v2f b3n = *(const v2f*)(br3 + k0);
        acc0 = WMMA_F32(a, b0, acc0);
        acc1 = WMMA_F32(a, b1, acc1);
        acc2 = WMMA_F32(a, b2, acc2);
        acc3 = WMMA_F32(a, b3, acc3);
        a = an; b0 = b0n; b1 = b1n; b2 = b2n; b3 = b3n;
    }
    // epilogue
    acc0 = WMMA_F32(a, b0, acc0);
    acc1 = WMMA_F32(a, b1, acc1);
    acc2 = WMMA_F32(a, b2, acc2);
    acc3 = WMMA_F32(a, b3, acc3);

    for (int j = 0; j < 8; ++j) {
        long r = (long)(m0 + j + 8 * hi) * N + n0 + lo;
        Cb[r]      = acc0[j];
        Cb[r + 16] = acc1[j];
        Cb[r + 32] = acc2[j];
        Cb[r + 48] = acc3[j];
    }
}

// Transpose small weight matrix: in [K,N] row-major -> out [N,K] row-major
__global__ void transpose_kernel(const float* __restrict__ in, float* __restrict__ out,
                                 int K, int N) {
    long i = (long)blockIdx.x * 256 + threadIdx.x;
    if (i >= (long)K * N) return;
    int n = (int)(i % N), k = (int)(i / N);
    out[(long)n * K + k] = in[i];
}

// ---------------------------------------------------------------------------
// Temporal decay weights: tw[b,s] = 0.95^(max_s ts - ts)
__global__ void tw_kernel(const float* __restrict__ ts, float* __restrict__ tw) {
    int b = blockIdx.x, tid = threadIdx.x;  // blockDim = 128
    __shared__ float sm[SEQ];
    float v = ts[b * SEQ + tid];
    sm[tid] = v; __syncthreads();
    for (int off = 64; off > 0; off >>= 1) {
        if (tid < off) sm[tid] = fmaxf(sm[tid], sm[tid + off]);
        __syncthreads();
    }
    tw[b * SEQ + tid] = powf(0.95f, sm[0] - v);
}

// Row L2 norms of sims: one wave per row of length SEQ
__global__ void rownorm_kernel(const float* __restrict__ sims, float* __restrict__ nrm) {
    int wave = threadIdx.x >> 5, lane = threadIdx.x & 31;
    long row = (long)blockIdx.x * 8 + wave;   // rows = BATCH*SEQ
    const float* p = sims + row * SEQ;
    float s = 0.f;
    for (int i = lane; i < SEQ; i += 32) { float v = p[i]; s += v * v; }
    for (int off = 16; off > 0; off >>= 1) s += __shfl_xor(s, off, 32);
    if (lane == 0) nrm[row] = fmaxf(sqrtf(s), 1e-12f);
}

// Edge MLP + threshold/triu mask: rewrites sims in place with scores (or -1)
__global__ void edge_kernel(float* __restrict__ sims, const float* __restrict__ nrm,
                            const float* __restrict__ tw, const float* __restrict__ iw,
                            const float* __restrict__ w1, const float* __restrict__ b1,
                            const float* __restrict__ w2, const float* __restrict__ b2) {
    __shared__ float sw1[48], sb1[16], sw2[16], sb2;
    int tid = threadIdx.x;
    if (tid < 48) sw1[tid] = w1[tid];
    if (tid < 16) { sb1[tid] = b1[tid]; sw2[tid] = w2[tid]; }
    if (tid == 0) sb2 = b2[0];
    __syncthreads();
    long i = (long)blockIdx.x * 256 + tid;        // over BATCH*SEQ*SEQ
    int b = (int)(i / (SEQ * SEQ));
    int r = (int)(i % (SEQ * SEQ));
    int s = r / SEQ, t = r % SEQ;
    float f0 = sims[i] / nrm[b * SEQ + s];
    float f1 = tw[b * SEQ + s] * tw[b * SEQ + t];
    float f2 = iw[b * SEQ + s];
    float acc = sb2;
    for (int j = 0; j < 16; ++j) {
        float h = fmaxf(f0 * sw1[j] + f1 * sw1[16 + j] + f2 * sw1[32 + j] + sb1[j], 0.f);
        acc += h * sw2[j];
    }
    float raw = 1.f / (1.f + expf(-acc));
    sims[i] = (t > s && raw > 0.1f) ? raw : -1.0f;
}

// Per-batch exact top-64 (iterative argmax, smaller-index tie-break), + self loops
__global__ void topk_kernel(float* __restrict__ score, int* __restrict__ srci,
                            int* __restrict__ dsti, float* __restrict__ emv) {
    int b = blockIdx.x, tid = threadIdx.x;     // blockDim = 256
    float* sc = score + (long)b * SEQ * SEQ;
    __shared__ float rv[256]; __shared__ int ri[256];
    for (int e = 0; e < KTOP; ++e) {
        float best = -3.f; int bi = 0;
        for (int i = tid; i < SEQ * SEQ; i += 256) {
            float v = sc[i];
            if (v > best) { best = v; bi = i; }
        }
        rv[tid] = best; ri[tid] = bi;
        __syncthreads();
        for (int off = 128; off > 0; off >>= 1) {
            if (tid < off) {
                float v2 = rv[tid + off]; int i2 = ri[tid + off];
                if (v2 > rv[tid] || (v2 == rv[tid] && i2 < ri[tid])) { rv[tid] = v2; ri[tid] = i2; }
            }
            __syncthreads();
        }
        if (tid == 0) {
            int idx = ri[0]; float v = rv[0];
            srci[b * EDGES + e] = idx / SEQ;
            dsti[b * EDGES + e] = idx % SEQ;
            emv [b * EDGES + e] = (v > 0.f) ? 1.f : 0.f;
            sc[idx] = -2.f;
            __threadfence_block();
        }
        __syncthreads();
    }
    for (int i = tid; i < SEQ; i += 256) {
        srci[b * EDGES + KTOP + i] = i;
        dsti[b * EDGES + KTOP + i] = i;
        emv [b * EDGES + KTOP + i] = 1.f;
    }
}

// Attention logits a_s/a_d : one thread per (b, s, h)
__global__ void att_kernel(const float* __restrict__ xp, const float* __restrict__ att_src,
                           const float* __restrict__ att_dst, float* __restrict__ a_s,
                           float* __restrict__ a_d, int fout, int fp, long total) {
    long i = (long)blockIdx.x * 256 + threadIdx.x;
    if (i >= total) return;
    int h = (int)(i % HEADS); long bs = i / HEADS;
    const float* row = xp + bs * fout + (long)h * fp;
    float s1 = 0.f, s2 = 0.f;
    for (int f = 0; f < fp; ++f) {
        s1 += row[f] * att_src[h * fp + f];
        s2 += row[f] * att_dst[h * fp + f];
    }
    a_s[i] = s1; a_d[i] = s2;
}

__global__ void zero_kernel(float* __restrict__ p, long n) {
    long i = (long)blockIdx.x * 256 + threadIdx.x;
    if (i < n) p[i] = 0.f;
}

__device__ inline void atomic_fmax_shared(float* addr, float val) {
    int* ia = (int*)addr;
    int cur = __float_as_int(*addr);
    while (__int_as_float(cur) < val) {
        int prev = atomicCAS(ia, cur, __float_as_int(val));
        if (prev == cur) break;
        cur = prev;
    }
}

// Per-batch GAT edge softmax + scatter. LDS: per-dst max (CAS fmax), edge weights.
// denom & feature accumulation go to global with native f32 atomic adds.
__global__ void scatter_kernel(const float* __restrict__ xp, const float* __restrict__ a_s,
                               const float* __restrict__ a_d, const int* __restrict__ srci,
                               const int* __restrict__ dsti, const float* __restrict__ emv,
                               float* __restrict__ accum, float* __restrict__ denom,
                               int fout, int fp) {
    int b = blockIdx.x, tid = threadIdx.x;   // blockDim = 256
    __shared__ float mx[SEQ * HEADS];        // 1024 f
    __shared__ float ea[EDGES * HEADS];      // 1536 f
    for (int i = tid; i < SEQ * HEADS; i += 256) mx[i] = -1e9f;
    __syncthreads();
    const int* sb = srci + b * EDGES;
    const int* db = dsti + b * EDGES;
    const float* mb = emv + b * EDGES;
    long base = (long)b * SEQ;
    for (int item = tid; item < EDGES * HEADS; item += 256) {
        int e = item / HEADS, h = item % HEADS;
        int se = sb[e], de = db[e];
        float v = a_s[(base + se) * HEADS + h] + a_d[(base + de) * HEADS + h];
        v = v > 0.f ? v : 0.2f * v;                 // leaky_relu
        if (mb[e] == 0.f) v = -1e9f;
        ea[item] = v;
        atomic_fmax_shared(&mx[de * HEADS + h], v);
    }
    __syncthreads();
    for (int item = tid; item < EDGES * HEADS; item += 256) {
        int e = item / HEADS, h = item % HEADS;
        int de = db[e];
        float a = expf(ea[item] - mx[de * HEADS + h]);
        ea[item] = a;
        atomicAdd(&denom[(base + de) * HEADS + h], a);
    }
    __syncthreads();
    long items = (long)EDGES * fout;
    for (long item = tid; item < items; item += 256) {
        int e = (int)(item / fout), f = (int)(item % fout);
        int h = f / fp;
        float a = ea[e * HEADS + h];
        if (a != 0.f) {
            int se = sb[e], de = db[e];
            atomicAdd(&accum[(base + de) * fout + f], a * xp[(base + se) * fout + f]);
        }
    }
}

// accum = accum / max(denom,1e-16) + bias
__global__ void finalize_kernel(float* __restrict__ accum, const float* __restrict__ denom,
                                const float* __restrict__ bias, int fout, int fp, long n) {
    long i = (long)blockIdx.x * 256 + threadIdx.x;
    if (i >= n) return;
    long bs = i / fout; int f = (int)(i % fout); int h = f / fp;
    accum[i] = accum[i] / fmaxf(denom[bs * HEADS + h], 1e-16f) + bias[f];
}

// LayerNorm(+res_b) + ReLU, one block per row, blockDim = fout (64/128/256)
__global__ void ln_kernel(const float* __restrict__ accum, const float* __restrict__ res_b,
                          const float* __restrict__ g, const float* __restrict__ bb,
                          float* __restrict__ out, int fout) {
    long row = blockIdx.x; int j = threadIdx.x;
    __shared__ float sm[256];
    float t = accum[row * fout + j] + res_b[j];
    sm[j] = t; __syncthreads();
    for (int off = fout >> 1; off > 0; off >>= 1) {
        if (j < off) sm[j] += sm[j + off];
        __syncthreads();
    }
    float mu = sm[0] / fout;
    __syncthreads();
    float d = t - mu;
    sm[j] = d * d; __syncthreads();
    for (int off = fout >> 1; off > 0; off >>= 1) {
        if (j < off) sm[j] += sm[j + off];
        __syncthreads();
    }
    float var = sm[0] / fout;
    float y = d * rsqrtf(var + 1e-5f) * g[j] + bb[j];
    out[row * fout + j] = fmaxf(y, 0.f);
}

// ---------------------------------------------------------------------------
extern "C" void kernel_launch(void* const* d_in, const int* in_sizes, int n_in,
                              void* d_out, int out_size, void* d_ws, size_t ws_size,
                              hipStream_t stream) {
    const float* items = (const float*)d_in[0];   // [B,S,D]
    const float* ts    = (const float*)d_in[1];   // [B,S]
    const float* iw    = (const float*)d_in[2];   // [B,S]
    const float* ew1   = (const float*)d_in[3];
    const float* eb1   = (const float*)d_in[4];
    const float* ew2   = (const float*)d_in[5];
    const float* eb2   = (const float*)d_in[6];
    // layers: d_in[7 + 8*l + {w,att_src,att_dst,bias,res_w,res_b,ln_g,ln_b}]

    const long BSS = (long)BATCH * SEQ * SEQ;     // 4.19M
    const long BSF = (long)BATCH * SEQ * 256;     // 8.39M (max fout)
    float* ws    = (float*)d_ws;
    float* sims  = ws;                 // BSS
    float* xp    = sims + BSS;         // BSF
    float* accum = xp + BSF;           // BSF
    float* bufA  = accum + BSF;        // BSF (layer0 output, 256 wide)
    float* bufB  = bufA + BSF;         // B*S*128
    float* tw    = bufB + (long)BATCH * SEQ * 128;
    float* nrm   = tw + (long)BATCH * SEQ;
    float* a_s   = nrm + (long)BATCH * SEQ;
    float* a_d   = a_s + (long)BATCH * SEQ * HEADS;
    float* denom = a_d + (long)BATCH * SEQ * HEADS;
    float* emv   = denom + (long)BATCH * SEQ * HEADS;
    int*   srci  = (int*)(emv + (long)BATCH * EDGES);
    int*   dsti  = srci + (long)BATCH * EDGES;
    float* wT    = (float*)(dsti + (long)BATCH * EDGES);   // 32768 (max fin*fout)
    float* rwT   = wT + 32768;                             // 32768

    // ---- graph construction ----
    tw_kernel<<<BATCH, SEQ, 0, stream>>>(ts, tw);
    gemm_nt_f32<<<dim3(SEQ / 64, SEQ / 16, BATCH), 32, 0, stream>>>(
        items, items, sims, SEQ, SEQ, DIM,
        (long)SEQ * DIM, (long)SEQ * DIM, (long)SEQ * SEQ, 0);
    rownorm_kernel<<<(BATCH * SEQ) / 8, 256, 0, stream>>>(sims, nrm);
    edge_kernel<<<(unsigned)(BSS / 256), 256, 0, stream>>>(sims, nrm, tw, iw,
                                                           ew1, eb1, ew2, eb2);
    topk_kernel<<<BATCH, 256, 0, stream>>>(sims, srci, dsti, emv);

    // ---- GAT layers ----
    const int FIN[3]  = {128, 256, 128};
    const int FOUT[3] = {256, 128, 64};
    const float* xin[3]  = {items, bufA, bufB};
    float*       xout[3] = {bufA, bufB, (float*)d_out};

    for (int l = 0; l < 3; ++l) {
        const float* w       = (const float*)d_in[7 + 8 * l + 0];
        const float* att_src = (const float*)d_in[7 + 8 * l + 1];
        const float* att_dst = (const float*)d_in[7 + 8 * l + 2];
        const float* bias    = (const float*)d_in[7 + 8 * l + 3];
        const float* res_w   = (const float*)d_in[7 + 8 * l + 4];
        const float* res_b   = (const float*)d_in[7 + 8 * l + 5];
        const float* ln_g    = (const float*)d_in[7 + 8 * l + 6];
        const float* ln_b    = (const float*)d_in[7 + 8 * l + 7];
        int fin = FIN[l], fout = FOUT[l], fp = fout / HEADS;
        long nBSF = (long)BATCH * SEQ * fout;
        long nBSH = (long)BATCH * SEQ * HEADS;
        int wcnt = fin * fout;

        // transpose weights to [fout, fin] so B fragments are contiguous b64 loads
        transpose_kernel<<<(wcnt + 255) / 256, 256, 0, stream>>>(w, wT, fin, fout);
        transpose_kernel<<<(wcnt + 255) / 256, 256, 0, stream>>>(res_w, rwT, fin, fout);

        // xp = x @ W  (WMMA, 16x64 tile per wave, pipelined)
        gemm_nt_f32<<<dim3(fout / 64, SEQ / 16, BATCH), 32, 0, stream>>>(
            xin[l], wT, xp, SEQ, fout, fin, (long)SEQ * fin, 0L, (long)SEQ * fout, 0);
        // attention logits
        att_kernel<<<(unsigned)((nBSH + 255) / 256), 256, 0, stream>>>(
            xp, att_src, att_dst, a_s, a_d, fout, fp, nBSH);
        // zero accumulators
        zero_kernel<<<(unsigned)((nBSF + 255) / 256), 256, 0, stream>>>(accum, nBSF);
        zero_kernel<<<(unsigned)((nBSH + 255) / 256), 256, 0, stream>>>(denom, nBSH);
        // edge softmax + scatter
        scatter_kernel<<<BATCH, 256, 0, stream>>>(xp, a_s, a_d, srci, dsti, emv,
                                                  accum, denom, fout, fp);
        // divide by denom, add bias
        finalize_kernel<<<(unsigned)((nBSF + 255) / 256), 256, 0, stream>>>(
            accum, denom, bias, fout, fp, nBSF);
        // accum += x @ res_w  (WMMA, beta=1)
        gemm_nt_f32<<<dim3(fout / 64, SEQ / 16, BATCH), 32, 0, stream>>>(
            xin[l], rwT, accum, SEQ, fout, fin, (long)SEQ * fin, 0L, (long)SEQ * fout, 1);
        // LayerNorm(+res_b) + ReLU
        ln_kernel<<<BATCH * SEQ, fout, 0, stream>>>(accum, res_b, ln_g, ln_b,
                                                    xout[l], fout);
    }
}